// FlowHistogram_30339648979573
// MI455X (gfx1250) — compile-verified
//
#include <hip/hip_runtime.h>
#include <math.h>

typedef __attribute__((ext_vector_type(2))) float v2f;
typedef __attribute__((ext_vector_type(8))) float v8f;

#define N_ROWT 8
#define TDIM   16
#define HDIM   512
#define WDIM   512
#define YSTR   65        // H//8 + 1
#define DEPTH  4
#define PI_F   3.14159265358979323846f

// Issue one row (x and y component) as async global->LDS b128 copies.
// Each of 128 threads copies 16B per component; per wave that is one 512B
// coalesced request per component, 2 ASYNCcnt increments per row per wave.
#define ISSUE(slot, rowi) do {                                              \
    uint32_t vo_ = (uint32_t)(((row0 + (rowi)) * WDIM + x0) * 4);           \
    asm volatile(                                                           \
        "global_load_async_to_lds_b128 %0, %2, %3\n\t"                      \
        "global_load_async_to_lds_b128 %1, %2, %4"                          \
        :: "v"(lx0 + (uint32_t)((slot) * (WDIM * 4))),                      \
           "v"(ly0 + (uint32_t)((slot) * (WDIM * 4))),                      \
           "v"(vo_), "s"(xg), "s"(yg)                                       \
        : "memory");                                                        \
} while (0)

__global__ __launch_bounds__(128)
void FlowHistogram_kernel(const float* __restrict__ flow, float* __restrict__ out) {
    const int tid = threadIdx.x;
    const int bt  = blockIdx.x >> 3;   // b*T + t  (0..127)
    const int r   = blockIdx.x & 7;    // tile row (0..7)
    const int b   = bt >> 4;
    const int t   = bt & 15;

    const size_t HW     = (size_t)HDIM * WDIM;
    const size_t planeX = ((size_t)(b * 2) * TDIM + t) * HW;       // flow[b,0,t]
    const size_t planeY = planeX + (size_t)TDIM * HW;              // flow[b,1,t]
    const uint64_t xg = (uint64_t)(uintptr_t)(flow + planeX);
    const uint64_t yg = (uint64_t)(uintptr_t)(flow + planeY);

    __shared__ alignas(16) float xbuf[DEPTH][WDIM];   // 8 KB
    __shared__ alignas(16) float ybuf[DEPTH][WDIM];   // 8 KB
    __shared__ float red[128][16];                    // 8 KB (cols 0..8 live)

    // LDS byte offsets for this thread's 16B staging slots (flat addr low 32 bits == LDS offset)
    const uint32_t lx0 = (uint32_t)(uintptr_t)&xbuf[0][0] + (uint32_t)tid * 16u;
    const uint32_t ly0 = (uint32_t)(uintptr_t)&ybuf[0][0] + (uint32_t)tid * 16u;

    const int rows = (r == 7) ? (HDIM - 7 * YSTR) : YSTR;   // 57 or 65
    const int row0 = r * YSTR;

    const int  x0 = tid * 4;                // first of 4 owned columns
    const int  c0 = x0 / YSTR;              // tile of col x0; cols span <= 2 tiles
    const bool m1 = ((x0 + 1) / YSTR) == c0;
    const bool m2 = ((x0 + 2) / YSTR) == c0;
    const bool m3 = ((x0 + 3) / YSTR) == c0;

    // ---- prologue: prefetch DEPTH rows into LDS ----
#pragma unroll
    for (int i = 0; i < DEPTH; ++i) ISSUE(i, i);

    float accA = 0.0f, accB = 0.0f;

    for (int row = 0; row < rows; ++row) {
        // wait until row `row` has landed (async ops complete in order per wave)
        int rem = rows - 1 - row; if (rem > DEPTH - 1) rem = DEPTH - 1;
        if (rem == 3)      asm volatile("s_wait_asynccnt 6" ::: "memory");
        else if (rem == 2) asm volatile("s_wait_asynccnt 4" ::: "memory");
        else if (rem == 1) asm volatile("s_wait_asynccnt 2" ::: "memory");
        else               asm volatile("s_wait_asynccnt 0" ::: "memory");

        const int slot = row & (DEPTH - 1);
        const float4 xv = *(const float4*)&xbuf[slot][x0];
        const float4 yv = *(const float4*)&ybuf[slot][x0];

        const float a0 = atan2f(yv.x, xv.x);
        const float a1 = atan2f(yv.y, xv.y);
        const float a2 = atan2f(yv.z, xv.z);
        const float a3 = atan2f(yv.w, xv.w);

        accA += a0;
        accA += m1 ? a1 : 0.0f;  accB += m1 ? 0.0f : a1;
        accA += m2 ? a2 : 0.0f;  accB += m2 ? 0.0f : a2;
        accA += m3 ? a3 : 0.0f;  accB += m3 ? 0.0f : a3;

        const int nxt = row + DEPTH;
        if (nxt < rows) ISSUE(nxt & (DEPTH - 1), nxt);
    }

    // ---- deposit per-thread partials into 128x16 matrix (zero padded) ----
#pragma unroll
    for (int i = 0; i < 16; ++i) red[tid][i] = 0.0f;
    red[tid][c0]     = accA;
    red[tid][c0 + 1] = accB;     // c0+1 <= 8 < 16; contributes 0 when unused

    // ---- tree reduce 128 rows -> 4 rows ----
    for (int s = 64; s >= 4; s >>= 1) {
        __syncthreads();
        for (int i = tid; i < s * 16; i += 128) {
            const int rr = i >> 4, cc = i & 15;
            red[rr][cc] += red[rr + s][cc];
        }
    }
    __syncthreads();

    // ---- final 4x16 column-sum via one f32 WMMA (wave 0, EXEC all ones) ----
    if (tid < 32) {
        const int lane = tid;
        const int colB = lane & 15;
        const int k0   = (lane < 16) ? 0 : 2;   // B: VGPR0 holds K=k0 row, VGPR1 K=k0+1

        v2f Av; Av.x = 1.0f;            Av.y = 1.0f;             // A = ones(16x4)
        v2f Bv; Bv.x = red[k0][colB];   Bv.y = red[k0 + 1][colB];
        v8f Cv = {};
        // D[m][n] = sum_k A[m][k]*B[k][n] = column sums of the 4x16 slab
        v8f Dv = __builtin_amdgcn_wmma_f32_16x16x4_f32(
            false, Av, false, Bv, (short)0, Cv, false, false);

        if (lane < 8) {
            const float sum  = Dv[0];                              // D[0][lane]
            const float rc   = (r    == 7) ? 57.0f : 65.0f;
            const float cc   = (lane == 7) ? 57.0f : 65.0f;
            const float mean = sum / (rc * cc);
            float q = floorf(8.0f * (PI_F + PI_F / 8.0f + mean) / (2.0f * PI_F));
            if (q == 8.0f) q = 0.0f;
            if (q < 0.0f)  q = -1.0f;
            out[((size_t)bt * 8 + r) * 8 + lane] = q;
        }
    }
}

extern "C" void kernel_launch(void* const* d_in, const int* in_sizes, int n_in,
                              void* d_out, int out_size, void* d_ws, size_t ws_size,
                              hipStream_t stream) {
    (void)in_sizes; (void)n_in; (void)out_size; (void)d_ws; (void)ws_size;
    const float* flow = (const float*)d_in[0];
    float* out = (float*)d_out;
    // 8 (B) * 16 (T) * 8 (tile rows) = 1024 blocks; 128 threads each
    FlowHistogram_kernel<<<dim3(1024), dim3(128), 0, stream>>>(flow, out);
}